// Pathway_GCN_8031588843578
// MI455X (gfx1250) — compile-verified
//
#include <hip/hip_runtime.h>

#define P_CNT 64
#define N_DIM 256
#define F_DIM 4096
#define H_DIM 512
#define O_DIM 64

typedef __attribute__((ext_vector_type(2))) float v2f;
typedef __attribute__((ext_vector_type(8))) float v8f;

// ---------------------------------------------------------------------------
// CDNA5 async copy: global -> LDS, 16 bytes per lane, tracked by ASYNCcnt.
// ---------------------------------------------------------------------------
__device__ __forceinline__ void async_ld16(float* lds_ptr, const float* gptr) {
  unsigned lds_addr = (unsigned)(unsigned long long)lds_ptr;
  asm volatile("global_load_async_to_lds_b128 %0, %1, off"
               :
               : "v"(lds_addr), "v"(gptr)
               : "memory");
}
#define ASYNC_WAIT(n) asm volatile("s_wait_asynccnt " #n ::: "memory")

// A/B fragment from a row-natural [k][col] LDS tile (row stride ld floats).
// VGPR0: lanes 0-15 -> K=0, lanes 16-31 -> K=2 ; VGPR1: K=1 / K=3.
__device__ __forceinline__ v2f frag_ld(const float* t, int ld, int lane) {
  const int c  = lane & 15;
  const int kk = (lane >> 4) << 1;  // 0 or 2
  v2f f;
  f.x = t[kk * ld + c];
  f.y = t[(kk + 1) * ld + c];
  return f;
}

// B fragment from a k-pair-interleaved tile: element (k,c) stored at
// ((k>>1)*ld + c)*2 + (k&1)  ->  one aligned 8-byte ds_load_b64 per fragment.
// t must point at float index ((k0>>1)*ld + c0)*2.
__device__ __forceinline__ v2f frag_ldi(const float* t, int ld, int lane) {
  const int c = lane & 15;
  const int g = lane >> 4;  // k-group +0 / +1 (covers k0..k0+3)
  const float2 q = *(const float2*)&t[(g * ld + c) * 2];
  v2f f;
  f.x = q.x;
  f.y = q.y;
  return f;
}

__device__ __forceinline__ v8f wmma4(v2f a, v2f b, v8f c) {
  return __builtin_amdgcn_wmma_f32_16x16x4_f32(false, a, false, b, (short)0, c,
                                               false, false);
}

// ---------------------------------------------------------------------------
// Kernel A:  W1p[p] = lpls[p]^T @ W1[p]   (256x256)^T x (256x512)
// Output stored CHUNKED+INTERLEAVED: per p, per 64-hidden chunk hc (contig
// 16384 floats):  offset = hc*16384 + (k>>1)*128 + hl*2 + (k&1).
// grid = (H/64, N/64, P), block = 256 threads (8 waves, 16x32 wave tiles)
// ---------------------------------------------------------------------------
__global__ __launch_bounds__(256) void w1prime_kernel(
    const float* __restrict__ lpls, const float* __restrict__ W1,
    float* __restrict__ W1p) {
  __shared__ __align__(16) float Lt[64 * 64];  // [k][m] = L[kb+k][m0+m]
  __shared__ __align__(16) float Wt[64 * 64];  // [k][n] = W1[kb+k][n0+n]

  const int p    = blockIdx.z;
  const int m0   = blockIdx.y * 64;
  const int n0   = blockIdx.x * 64;  // hidden chunk base (chunk = n0>>6)
  const int tid  = threadIdx.x;
  const int lane = tid & 31;
  const int wave = tid >> 5;
  const int wm   = (wave & 3) * 16;
  const int wn   = (wave >> 2) * 32;

  const float* Lp  = lpls + (size_t)p * N_DIM * N_DIM;
  const float* W1g = W1 + (size_t)p * N_DIM * H_DIM;

  v8f acc0 = {};
  v8f acc1 = {};

  for (int kb = 0; kb < N_DIM; kb += 64) {
    for (int i = tid * 4; i < 64 * 64; i += 256 * 4) {
      const int k = i >> 6, x = i & 63;
      async_ld16(&Lt[i], &Lp[(size_t)(kb + k) * N_DIM + m0 + x]);
      async_ld16(&Wt[i], &W1g[(size_t)(kb + k) * H_DIM + n0 + x]);
    }
    ASYNC_WAIT(0);
    __syncthreads();
#pragma unroll
    for (int k = 0; k < 64; k += 4) {
      const v2f a  = frag_ld(&Lt[k * 64 + wm], 64, lane);
      const v2f b0 = frag_ld(&Wt[k * 64 + wn], 64, lane);
      const v2f b1 = frag_ld(&Wt[k * 64 + wn + 16], 64, lane);
      acc0 = wmma4(a, b0, acc0);
      acc1 = wmma4(a, b1, acc1);
    }
    __syncthreads();
  }

  // Store 16x32 wave tile into chunked+interleaved W1p.
  float* Og = W1p + (size_t)p * N_DIM * H_DIM + (size_t)(n0 >> 6) * (N_DIM * 64);
  const int n   = lane & 15;
  const int mh8 = (lane >> 4) * 8;
#pragma unroll
  for (int r = 0; r < 8; ++r) {
    const int m    = m0 + wm + mh8 + r;  // k index of W1p
    const int base = (m >> 1) * 128 + (m & 1);
    Og[base + (wn + n) * 2]      = acc0[r];
    Og[base + (wn + 16 + n) * 2] = acc1[r];
  }
}

// ---------------------------------------------------------------------------
// Kernel C: repack W2 into chunked+interleaved layout (8.4 MB, one pass):
//   W2r[p][kc*4096 + (kl>>1)*128 + o*2 + (kl&1)] = W2[p][k][o]
// ---------------------------------------------------------------------------
__global__ __launch_bounds__(256) void w2repack_kernel(
    const float* __restrict__ W2, float* __restrict__ W2r) {
  const int idx = blockIdx.x * 256 + threadIdx.x;  // over P*H*O
  const int o  = idx & 63;
  const int k  = (idx >> 6) & (H_DIM - 1);
  const int p  = idx >> 15;
  const int kc = k >> 6, kl = k & 63;
  W2r[(size_t)p * H_DIM * O_DIM + kc * 4096 + (kl >> 1) * 128 + o * 2 +
      (kl & 1)] = W2[idx];
}

// ---------------------------------------------------------------------------
// Kernel B: fused MLP per (pathway, 128-cell tile):
//   H = relu(E^T @ W1p + b1) ; Out = H @ W2 + b2, H never leaves LDS.
// grid = (F/128, P), block = 256 threads (8 waves, 32x32 wave tiles),
// double-buffered async W1' prefetch; B operands k-pair interleaved.
// ---------------------------------------------------------------------------
__global__ __launch_bounds__(256) void mlp_fused_kernel(
    const float* __restrict__ expg, const float* __restrict__ W1p,
    const float* __restrict__ b1g, const float* __restrict__ W2r,
    const float* __restrict__ b2g, float* __restrict__ out) {
  __shared__ __align__(16) float At[N_DIM * 128];     // [n=256][f=128] 128 KB
  __shared__ __align__(16) float B1t[2][N_DIM * 64];  // interleaved, x2 128 KB
  __shared__ __align__(16) float Ht[64 * 129];        // [h=64][f=128]+pad 33 KB
  __shared__ __align__(16) float B2t[64 * 64];        // interleaved     16 KB

  const int p    = blockIdx.y;
  const int f0   = blockIdx.x * 128;
  const int tid  = threadIdx.x;
  const int lane = tid & 31;
  const int wave = tid >> 5;
  const int wf   = (wave & 3) * 32;   // wave f offset (0..96)
  const int wn   = (wave >> 2) * 32;  // wave n offset (h or o) (0/32)

  const float* Ep  = expg + (size_t)p * N_DIM * F_DIM;
  const float* W1g = W1p + (size_t)p * N_DIM * H_DIM;  // chunked+interleaved
  const float* W2p = W2r + (size_t)p * H_DIM * O_DIM;  // chunked+interleaved
  const float* b1p = b1g + (size_t)p * H_DIM;
  const float* b2p = b2g + (size_t)p * O_DIM;

  // Prologue: async-stage E^T tile (32 instr/wave) + W1' chunk 0 (16/wave).
  for (int i = tid * 4; i < N_DIM * 128; i += 256 * 4) {
    const int n = i >> 7, f = i & 127;
    async_ld16(&At[i], &Ep[(size_t)n * F_DIM + f0 + f]);
  }
  for (int i = tid * 4; i < N_DIM * 64; i += 256 * 4) {
    async_ld16(&B1t[0][i], &W1g[i]);  // chunk 0 is contiguous
  }

  v8f o00 = {}, o01 = {}, o10 = {}, o11 = {};
  const int nl  = lane & 15;
  const int mh8 = (lane >> 4) * 8;

  for (int nb = 0; nb < H_DIM / 64; ++nb) {
    const int cur = nb & 1;
    const int hb0 = nb * 64;

    // Stage W2 chunk (4 instr/wave), then prefetch next W1' chunk (16/wave).
    // Async loads complete in order -> partial waits below.
    const float* W2c = W2p + (size_t)nb * 4096;
    for (int i = tid * 4; i < 64 * 64; i += 256 * 4) {
      async_ld16(&B2t[i], &W2c[i]);
    }
    if (nb + 1 < H_DIM / 64) {
      const float* W1c = W1g + (size_t)(nb + 1) * (N_DIM * 64);
      for (int i = tid * 4; i < N_DIM * 64; i += 256 * 4) {
        async_ld16(&B1t[cur ^ 1][i], &W1c[i]);
      }
      ASYNC_WAIT(20);  // all but B2t(4)+B1t_next(16) done => B1t[cur],At ready
    } else {
      ASYNC_WAIT(4);   // all but B2t(4) done
    }
    __syncthreads();

    // Pass 1: H chunk (128f x 64h) = E^T tile @ W1' chunk, K = 256.
    const float* B1c = B1t[cur];
    v8f h00 = {}, h01 = {}, h10 = {}, h11 = {};
#pragma unroll 4
    for (int k = 0; k < N_DIM; k += 4) {
      const v2f a0 = frag_ld(&At[k * 128 + wf], 128, lane);
      const v2f a1 = frag_ld(&At[k * 128 + wf + 16], 128, lane);
      const v2f b0 = frag_ldi(&B1c[(k >> 1) * 128 + wn * 2], 64, lane);
      const v2f b1 = frag_ldi(&B1c[(k >> 1) * 128 + (wn + 16) * 2], 64, lane);
      h00 = wmma4(a0, b0, h00);
      h01 = wmma4(a0, b1, h01);
      h10 = wmma4(a1, b0, h10);
      h11 = wmma4(a1, b1, h11);
    }

    // bias + relu, transposed into Ht[h][f] (stride 129 -> conflict-free).
    const float bias0 = b1p[hb0 + wn + nl];
    const float bias1 = b1p[hb0 + wn + 16 + nl];
#pragma unroll
    for (int r = 0; r < 8; ++r) {
      const int fa = wf + mh8 + r;
      Ht[(wn + nl) * 129 + fa]           = fmaxf(h00[r] + bias0, 0.0f);
      Ht[(wn + 16 + nl) * 129 + fa]      = fmaxf(h01[r] + bias1, 0.0f);
      Ht[(wn + nl) * 129 + fa + 16]      = fmaxf(h10[r] + bias0, 0.0f);
      Ht[(wn + 16 + nl) * 129 + fa + 16] = fmaxf(h11[r] + bias1, 0.0f);
    }
    if (nb + 1 < H_DIM / 64) {
      ASYNC_WAIT(16);  // B2t done; only next W1' prefetch may remain in flight
    } else {
      ASYNC_WAIT(0);
    }
    __syncthreads();

    // Pass 2: Out (128f x 64o) += H chunk @ W2 chunk, K = 64.
#pragma unroll 4
    for (int k = 0; k < 64; k += 4) {
      const v2f a0 = frag_ld(&Ht[k * 129 + wf], 129, lane);
      const v2f a1 = frag_ld(&Ht[k * 129 + wf + 16], 129, lane);
      const v2f b0 = frag_ldi(&B2t[(k >> 1) * 128 + wn * 2], 64, lane);
      const v2f b1 = frag_ldi(&B2t[(k >> 1) * 128 + (wn + 16) * 2], 64, lane);
      o00 = wmma4(a0, b0, o00);
      o01 = wmma4(a0, b1, o01);
      o10 = wmma4(a1, b0, o10);
      o11 = wmma4(a1, b1, o11);
    }
    __syncthreads();  // protect Ht/B2t (and B1' buffers) before next stage
  }

  // Epilogue: + b2, store out[p][f][o] (32f x 32o per wave).
  const float bb0 = b2p[wn + nl];
  const float bb1 = b2p[wn + 16 + nl];
#pragma unroll
  for (int r = 0; r < 8; ++r) {
    const size_t fA = (size_t)p * F_DIM + f0 + wf + mh8 + r;
    const size_t fB = fA + 16;
    out[fA * O_DIM + wn + nl]      = o00[r] + bb0;
    out[fA * O_DIM + wn + 16 + nl] = o01[r] + bb1;
    out[fB * O_DIM + wn + nl]      = o10[r] + bb0;
    out[fB * O_DIM + wn + 16 + nl] = o11[r] + bb1;
  }
}

extern "C" void kernel_launch(void* const* d_in, const int* in_sizes, int n_in,
                              void* d_out, int out_size, void* d_ws,
                              size_t ws_size, hipStream_t stream) {
  const float* lpls = (const float*)d_in[0];
  const float* expg = (const float*)d_in[1];
  const float* W1   = (const float*)d_in[2];
  const float* b1   = (const float*)d_in[3];
  const float* W2   = (const float*)d_in[4];
  const float* b2   = (const float*)d_in[5];
  float* out = (float*)d_out;
  float* W1p = (float*)d_ws;                          // 33.6 MB
  float* W2r = W1p + (size_t)P_CNT * N_DIM * H_DIM;   // + 8.4 MB

  dim3 gA(H_DIM / 64, N_DIM / 64, P_CNT);
  w1prime_kernel<<<gA, 256, 0, stream>>>(lpls, W1, W1p);

  w2repack_kernel<<<(P_CNT * H_DIM * O_DIM) / 256, 256, 0, stream>>>(W2, W2r);

  dim3 gB(F_DIM / 128, P_CNT);
  mlp_fused_kernel<<<gB, 256, 0, stream>>>(expg, W1p, b1, W2r, b2, out);
}